// HGCNLayer_general_4252017623766
// MI455X (gfx1250) — compile-verified
//
#include <hip/hip_runtime.h>
#include <hip/hip_bf16.h>
#include <stdint.h>

// HGCN layer: edge-wise gather + scatter-add with degree normalization.
// Memory-bound (~0.25 FLOP/byte). Strategy:
//  - hardware global_atomic_add_f32 (no-return) for all scatter-adds; the
//    full working set (~140MB) is resident in MI455X's 192MB L2.
//  - CDNA5 async DMA (global_load_async_to_lds_b128, ASYNCcnt) to
//    double-buffer each edge's 512B feature row into LDS, hiding random
//    gather latency behind the atomic scatter of the previous edge.
//  - wave32: one edge row = 128 f32 = one float4 per lane.

#define TPB 256
#define WPB 8   // waves per 256-thread block (wave32)

__device__ __forceinline__ void atomAddF(float* p, float v) {
    // Forces hardware FP32 global atomic (no CAS loop), no-return form.
    unsafeAtomicAdd(p, v);
}

__global__ void fill0_kernel(float* __restrict__ p, size_t n) {
    size_t i = (size_t)blockIdx.x * blockDim.x + threadIdx.x;
    size_t stride = (size_t)gridDim.x * blockDim.x;
    for (; i < n; i += stride) p[i] = 0.0f;
}

__global__ void degree_kernel(const int* __restrict__ ui_src, const int* __restrict__ ui_dst,
                              const int* __restrict__ ut_src, const int* __restrict__ ut_dst,
                              float* __restrict__ deg_i, float* __restrict__ deg_t,
                              float* __restrict__ degu_ui, float* __restrict__ degu_ut,
                              int eui, int eut) {
    int e = blockIdx.x * blockDim.x + threadIdx.x;
    if (e < eui) {
        atomAddF(&deg_i[ui_dst[e]], 1.0f);
        atomAddF(&degu_ui[ui_src[e]], 1.0f);
    } else if (e < eui + eut) {
        int f = e - eui;
        atomAddF(&deg_t[ut_dst[f]], 1.0f);
        atomAddF(&degu_ut[ut_src[f]], 1.0f);
    }
}

// accum[sidx[e]] += feat[gidx[e]]   for each edge e, rows of 128 f32.
// One wave per edge; async-DMA double buffer in LDS (2 x 512B per wave).
__global__ void scatter_kernel(const float* __restrict__ feat,
                               const int* __restrict__ gidx,
                               const int* __restrict__ sidx,
                               float* __restrict__ accum,
                               int nE) {
    __shared__ float buf[WPB][2][128];
    const int lane = threadIdx.x & 31;
    const int w = threadIdx.x >> 5;
    long long e = (long long)blockIdx.x * WPB + w;
    const long long stride = (long long)gridDim.x * WPB;
    if (e >= nE) return;   // wave-uniform exit: EXEC stays all-ones for async ops

    unsigned lds0 = (unsigned)(uintptr_t)(&buf[w][0][lane * 4]);
    unsigned lds1 = (unsigned)(uintptr_t)(&buf[w][1][lane * 4]);

    // Prologue: kick off first gather (each lane DMAs its 16B slice).
    {
        int s = gidx[e];
        unsigned long long g =
            (unsigned long long)(uintptr_t)(feat + (size_t)s * 128 + (size_t)lane * 4);
        asm volatile("global_load_async_to_lds_b128 %0, %1, off"
                     :: "v"(lds0), "v"(g) : "memory");
    }

    int parity = 0;
    while (true) {
        long long en = e + stride;
        if (en < nE) {
            // Prefetch next edge's row into the other buffer, then wait for
            // the current buffer (in-order ASYNCcnt: 2 outstanding -> wait 1).
            int s = gidx[en];
            unsigned long long g =
                (unsigned long long)(uintptr_t)(feat + (size_t)s * 128 + (size_t)lane * 4);
            unsigned ldsn = parity ? lds0 : lds1;
            asm volatile("global_load_async_to_lds_b128 %0, %1, off"
                         :: "v"(ldsn), "v"(g) : "memory");
            asm volatile("s_wait_asynccnt 1" ::: "memory");
        } else {
            asm volatile("s_wait_asynccnt 0" ::: "memory");
        }

        float4 v = *reinterpret_cast<const float4*>(&buf[w][parity][lane * 4]);
        int d = sidx[e];
        float* a = accum + (size_t)d * 128 + (size_t)lane * 4;
        atomAddF(a + 0, v.x);
        atomAddF(a + 1, v.y);
        atomAddF(a + 2, v.z);
        atomAddF(a + 3, v.w);

        e = en;
        if (e >= nE) break;
        parity ^= 1;
    }
}

// feat[r][j] /= max(deg[r], 1)
__global__ void norm_kernel(float* __restrict__ feat, const float* __restrict__ deg, size_t rows) {
    size_t i = (size_t)blockIdx.x * blockDim.x + threadIdx.x;
    if (i < rows * 128) {
        size_t r = i >> 7;
        feat[i] = feat[i] / fmaxf(deg[r], 1.0f);
    }
}

// b[r][j] /= (max(dui[r],1) + max(dut[r],1))
__global__ void scaleb_kernel(float* __restrict__ b, const float* __restrict__ dui,
                              const float* __restrict__ dut, size_t nu) {
    size_t i = (size_t)blockIdx.x * blockDim.x + threadIdx.x;
    if (i < nu * 128) {
        size_t r = i >> 7;
        b[i] = b[i] / (fmaxf(dui[r], 1.0f) + fmaxf(dut[r], 1.0f));
    }
}

static inline int scatter_grid(int nE) {
    long long b = ((long long)nE + WPB - 1) / WPB;
    if (b > 26624) b = 26624;  // grid-stride beyond this
    if (b < 1) b = 1;
    return (int)b;
}

extern "C" void kernel_launch(void* const* d_in, const int* in_sizes, int n_in,
                              void* d_out, int out_size, void* d_ws, size_t ws_size,
                              hipStream_t stream) {
    const int D = 128;
    const float* h_u   = (const float*)d_in[0];
    // d_in[1] (h_i), d_in[2] (h_t) are unused by the reference math.
    const int* ui_src  = (const int*)d_in[3];
    const int* ui_dst  = (const int*)d_in[4];
    const int* ut_src  = (const int*)d_in[5];
    const int* ut_dst  = (const int*)d_in[6];

    const size_t NU = (size_t)in_sizes[0] / D;
    const size_t NI = (size_t)in_sizes[1] / D;
    const size_t NT = (size_t)in_sizes[2] / D;
    const int EUI = in_sizes[3];
    const int EUT = in_sizes[5];

    // Output layout: [ bsrc (NU*D) | rst_t (NT*D) ]
    float* out   = (float*)d_out;
    float* bsrc  = out;
    float* rst_t = out + NU * D;   // accumulated + normalized in place

    // Workspace layout (floats): rst_i | deg_i | deg_t | degu_ui | degu_ut
    float* ws      = (float*)d_ws;
    float* rst_i   = ws;
    float* deg_i   = rst_i + NI * D;
    float* deg_t   = deg_i + NI;
    float* degu_ui = deg_t + NT;
    float* degu_ut = degu_ui + NU;
    const size_t ws_floats = NI * D + NI + NT + 2 * NU;

    // 1) zero accumulators (d_out + ws) — required every call (harness poisons)
    fill0_kernel<<<4096, TPB, 0, stream>>>(ws, ws_floats);
    fill0_kernel<<<4096, TPB, 0, stream>>>(out, (NU + NT) * D);

    // 2) per-etype degree histograms
    int ne = EUI + EUT;
    degree_kernel<<<(ne + TPB - 1) / TPB, TPB, 0, stream>>>(
        ui_src, ui_dst, ut_src, ut_dst, deg_i, deg_t, degu_ui, degu_ut, EUI, EUT);

    // 3) forward: rst_i += h_u[ui_src] @ ui_dst ; rst_t += h_u[ut_src] @ ut_dst
    scatter_kernel<<<scatter_grid(EUI), TPB, 0, stream>>>(h_u, ui_src, ui_dst, rst_i, EUI);
    scatter_kernel<<<scatter_grid(EUT), TPB, 0, stream>>>(h_u, ut_src, ut_dst, rst_t, EUT);

    // 4) normalize by clamped in-degree
    norm_kernel<<<(int)((NI * D + TPB - 1) / TPB), TPB, 0, stream>>>(rst_i, deg_i, NI);
    norm_kernel<<<(int)((NT * D + TPB - 1) / TPB), TPB, 0, stream>>>(rst_t, deg_t, NT);

    // 5) backward: bsrc += rst_i[ui_dst] @ ui_src  +  rst_t[ut_dst] @ ut_src
    scatter_kernel<<<scatter_grid(EUI), TPB, 0, stream>>>(rst_i, ui_dst, ui_src, bsrc, EUI);
    scatter_kernel<<<scatter_grid(EUT), TPB, 0, stream>>>(rst_t, ut_dst, ut_src, bsrc, EUT);

    // 6) bsrc *= 1 / (clamp(deg_ui_src) + clamp(deg_ut_src))
    scaleb_kernel<<<(int)((NU * D + TPB - 1) / TPB), TPB, 0, stream>>>(bsrc, degu_ui, degu_ut, NU);
}